// MPNN_Property_42099269435465
// MI455X (gfx1250) — compile-verified
//
#include <hip/hip_runtime.h>
#include <math.h>

// Problem constants (match reference setup_inputs)
#define N_NODES 100000
#define N_EDGES 1600000
#define NGRAPH 1000
#define H 64
#define FN 9
#define FE 3
#define FIN 12          // FN + FE
#define NUM_STEPS 6
#define THREE_H 192

typedef __attribute__((ext_vector_type(2))) float v2f;
typedef __attribute__((ext_vector_type(8))) float v8f;

// D = A(16x4,f32) * B(4x16,f32) + C(16x16,f32)   -- CDNA5 V_WMMA_F32_16X16X4_F32
__device__ __forceinline__ v8f wmma_k4(v2f a, v2f b, v8f c) {
  return __builtin_amdgcn_wmma_f32_16x16x4_f32(
      /*neg_a=*/false, a, /*neg_b=*/false, b,
      /*c_mod=*/(short)0, c, /*reuse_a=*/false, /*reuse_b=*/false);
}

__device__ __forceinline__ float sigmoidf_(float v) {
  return 1.0f / (1.0f + __expf(-v));
}

// ---------------------------------------------------------------------------
// Kernel 0: zero scratch (agg + pooled sums + counts are contiguous in ws)
// ---------------------------------------------------------------------------
__global__ void k_zero(float* __restrict__ p, long n) {
  long i = (long)blockIdx.x * blockDim.x + threadIdx.x;
  long stride = (long)gridDim.x * blockDim.x;
  for (; i < n; i += stride) p[i] = 0.0f;
}

// ---------------------------------------------------------------------------
// Kernel 0b: repack W[3H][H] (row-major) into k-pair-major layout
//   WT[(k>>1)*384 + n*2 + (k&1)] = W[n*64 + k]
// so a B-fragment (k,k+1 for column n) is one coalesced b64 load per lane.
// ---------------------------------------------------------------------------
__global__ void k_repack(const float* __restrict__ W, float* __restrict__ WT) {
  int i = blockIdx.x * blockDim.x + threadIdx.x;      // over 3H*H
  if (i < THREE_H * H) {
    int n = i >> 6, k = i & 63;
    WT[(k >> 1) * (2 * THREE_H) + n * 2 + (k & 1)] = W[i];
  }
}

// ---------------------------------------------------------------------------
// Kernel 1: fused edge message MLP (Linear(12->64)+ReLU, Linear(64->64) via
// WMMA) + atomic scatter-add into aggregated[N,H].
// Block = 256 threads = 8 waves; each wave owns exactly one 16-edge tile.
// E = 1.6M = 100000 tiles -> grid 12500, no tails.
// ---------------------------------------------------------------------------
#define SW2 68   // padded LDS stride: halves land in disjoint bank classes
#define SH1 68

__global__ __launch_bounds__(256) void k_edge_mlp(
    const float* __restrict__ x, const int* __restrict__ edge_index,
    const float* __restrict__ ea,
    const float* __restrict__ W1, const float* __restrict__ b1,
    const float* __restrict__ W2, const float* __restrict__ b2,
    float* __restrict__ agg)
{
  __shared__ float sW1[H * FIN];
  __shared__ float sb1[H];
  __shared__ float sW2[H * SW2];
  __shared__ float sb2[H];
  __shared__ float sMin[8][16 * FIN];
  __shared__ float sH1[8][16 * SH1];
  __shared__ int   sSrc[8][16];
  __shared__ int   sDst[8][16];

  const int tid = threadIdx.x;
  const int w   = tid >> 5;
  const int l   = tid & 31;

  for (int i = tid; i < H * FIN; i += 256) sW1[i] = W1[i];
  for (int i = tid; i < H; i += 256) { sb1[i] = b1[i]; sb2[i] = b2[i]; }
  for (int i = tid; i < H * H; i += 256)
    sW2[(i >> 6) * SW2 + (i & 63)] = W2[i];
  __syncthreads();

  const long e0 = ((long)blockIdx.x * 8 + w) * 16;
  if (l < 16) {
    sSrc[w][l] = edge_index[e0 + l];                 // row 0: src
    sDst[w][l] = edge_index[(long)N_EDGES + e0 + l]; // row 1: dst
  }
  __syncthreads();

  // stage m_in = concat(x[src], edge_attr) : 16 x 12
  for (int v = l; v < 16 * FIN; v += 32) {
    int m = v / FIN, i = v - m * FIN;
    float val;
    if (i < FN) val = x[(long)sSrc[w][m] * FN + i];
    else        val = ea[(e0 + m) * FE + (i - FN)];
    sMin[w][m * FIN + i] = val;
  }
  __syncthreads();

  // layer 1 + ReLU: lane handles edge m=l&15, output k-range (l>>4)*32..+32
  {
    const int m = l & 15;
    const int kb = (l >> 4) * 32;
    float in[FIN];
    #pragma unroll
    for (int i = 0; i < FIN; ++i) in[i] = sMin[w][m * FIN + i];
    #pragma unroll 4
    for (int kk = 0; kk < 32; ++kk) {
      int k = kb + kk;
      float acc = sb1[k];
      #pragma unroll
      for (int i = 0; i < FIN; ++i) acc += in[i] * sW1[k * FIN + i];
      sH1[w][m * SH1 + k] = fmaxf(acc, 0.0f);
    }
  }
  __syncthreads();

  // layer 2: messages(16x64) = h1(16x64) @ W2^T via WMMA.
  // k-step outer / nt inner -> 4 independent accumulator chains + A reuse.
  const int ma   = l & 15;        // A-frag: M index (lanes 16-31 mirror 0-15)
  const int koff = (l >> 4) * 2;  // A/B-frag: K sub-offset per half-wave
  v8f cc[4];
  #pragma unroll
  for (int nt = 0; nt < 4; ++nt)
    #pragma unroll
    for (int r = 0; r < 8; ++r) cc[nt][r] = 0.0f;

  #pragma unroll
  for (int ks = 0; ks < 16; ++ks) {
    const int k = ks * 4 + koff;
    v2f a;
    a.x = sH1[w][ma * SH1 + k];
    a.y = sH1[w][ma * SH1 + k + 1];
    #pragma unroll
    for (int nt = 0; nt < 4; ++nt) {
      const int n = nt * 16 + ma;
      v2f b;
      b.x = sW2[n * SW2 + k];          // B[k][n] = W2[n][k]
      b.y = sW2[n * SW2 + k + 1];
      cc[nt] = wmma_k4(a, b, cc[nt]);
    }
  }

  // D layout: lane l, vgpr r -> (m = r + 8*(l>>4), n = nt*16 + (l&15))
  const int mhalf = (l >> 4) * 8;
  #pragma unroll
  for (int nt = 0; nt < 4; ++nt) {
    const int nloc  = nt * 16 + (l & 15);
    const float bias = sb2[nloc];
    #pragma unroll
    for (int r = 0; r < 8; ++r) {
      int m = r + mhalf;
      atomicAdd(&agg[(long)sDst[w][m] * H + nloc], cc[nt][r] + bias);
    }
  }
}

// ---------------------------------------------------------------------------
// Kernel 2: gi = agg @ Wih^T + bih (WMMA, 12 interleaved chains), then 6
// fused GRU steps (gh = h @ Whh^T via WMMA in jt-pairs = 6 interleaved
// chains; gates elementwise in D-fragment layout), then global_mean_pool.
// Block = 256 threads = 8 waves x 16-node tiles = 128 nodes/block.
// WihT/WhhT are the k-pair-major repacked weights (coalesced B loads).
// ---------------------------------------------------------------------------
#define ST 68

__global__ __launch_bounds__(256) void k_gru_pool(
    const float* __restrict__ agg, const int* __restrict__ batch,
    const float* __restrict__ WihT, const float* __restrict__ WhhT,
    const float* __restrict__ bih, const float* __restrict__ bhh,
    float* __restrict__ sums, float* __restrict__ counts)
{
  __shared__ float sT[8][16 * ST];  // agg tile, then reused as h tile

  const int tid = threadIdx.x;
  const int w   = tid >> 5;
  const int l   = tid & 31;
  const int nodeBase = blockIdx.x * 128 + w * 16;

  // stage agg tile (zero-pad past N)
  for (int v = l; v < 16 * H; v += 32) {
    int m = v >> 6, k = v & 63;
    int node = nodeBase + m;
    sT[w][m * ST + k] = (node < N_NODES) ? agg[(long)node * H + k] : 0.0f;
  }
  __syncthreads();

  const int ma   = l & 15;
  const int koff = (l >> 4) * 2;
  const int ncol = l & 15;

  // gi: 12 D-tiles (16 nodes x 192 cols) in registers, bias pre-loaded.
  v8f gi[12];
  #pragma unroll
  for (int c = 0; c < 12; ++c) {
    float bias = bih[c * 16 + ncol];
    #pragma unroll
    for (int r = 0; r < 8; ++r) gi[c][r] = bias;
  }
  #pragma unroll
  for (int ks = 0; ks < 16; ++ks) {
    const int k = ks * 4 + koff;              // always even
    v2f a;
    a.x = sT[w][ma * ST + k];
    a.y = sT[w][ma * ST + k + 1];
    const long krow = (long)(k >> 1) * (2 * THREE_H);
    #pragma unroll
    for (int c = 0; c < 12; ++c) {
      const int ng = c * 16 + ncol;
      v2f b = *(const v2f*)&WihT[krow + ng * 2];   // (k,ng),(k+1,ng)
      gi[c] = wmma_k4(a, b, gi[c]);
    }
  }
  __syncthreads();

  // h = 0 (registers, fragment layout) and in LDS (A-operand layout)
  v8f hreg[4];
  #pragma unroll
  for (int jt = 0; jt < 4; ++jt)
    #pragma unroll
    for (int r = 0; r < 8; ++r) hreg[jt][r] = 0.0f;
  for (int v = l; v < 16 * H; v += 32) {
    int m = v >> 6, k = v & 63;
    sT[w][m * ST + k] = 0.0f;
  }
  __syncthreads();

  for (int step = 0; step < NUM_STEPS; ++step) {
    #pragma unroll
    for (int p = 0; p < 2; ++p) {             // jt-pairs {0,1}, {2,3}
      const int j0 = p * 2, j1 = p * 2 + 1;
      const int nr0 = j0 * 16 + ncol, nz0 = (j0 + 4) * 16 + ncol, nn0 = (j0 + 8) * 16 + ncol;
      const int nr1 = j1 * 16 + ncol, nz1 = (j1 + 4) * 16 + ncol, nn1 = (j1 + 8) * 16 + ncol;
      v8f gr0, gz0, gn0, gr1, gz1, gn1;
      {
        float a0 = bhh[nr0], a1 = bhh[nz0], a2 = bhh[nn0];
        float a3 = bhh[nr1], a4 = bhh[nz1], a5 = bhh[nn1];
        #pragma unroll
        for (int r = 0; r < 8; ++r) {
          gr0[r] = a0; gz0[r] = a1; gn0[r] = a2;
          gr1[r] = a3; gz1[r] = a4; gn1[r] = a5;
        }
      }
      #pragma unroll
      for (int ks = 0; ks < 16; ++ks) {
        const int k = ks * 4 + koff;
        v2f a;
        a.x = sT[w][ma * ST + k];             // old h (LDS untouched this step)
        a.y = sT[w][ma * ST + k + 1];
        const long krow = (long)(k >> 1) * (2 * THREE_H);
        v2f b;
        b = *(const v2f*)&WhhT[krow + nr0 * 2]; gr0 = wmma_k4(a, b, gr0);
        b = *(const v2f*)&WhhT[krow + nz0 * 2]; gz0 = wmma_k4(a, b, gz0);
        b = *(const v2f*)&WhhT[krow + nn0 * 2]; gn0 = wmma_k4(a, b, gn0);
        b = *(const v2f*)&WhhT[krow + nr1 * 2]; gr1 = wmma_k4(a, b, gr1);
        b = *(const v2f*)&WhhT[krow + nz1 * 2]; gz1 = wmma_k4(a, b, gz1);
        b = *(const v2f*)&WhhT[krow + nn1 * 2]; gn1 = wmma_k4(a, b, gn1);
      }
      // gates: gi, gh, h all share the same (m,n) slot per lane/vgpr
      #pragma unroll
      for (int r = 0; r < 8; ++r) {
        float rr = sigmoidf_(gi[j0][r] + gr0[r]);
        float zz = sigmoidf_(gi[j0 + 4][r] + gz0[r]);
        float nv = tanhf(gi[j0 + 8][r] + rr * gn0[r]);
        hreg[j0][r] = (1.0f - zz) * nv + zz * hreg[j0][r];
        rr = sigmoidf_(gi[j1][r] + gr1[r]);
        zz = sigmoidf_(gi[j1 + 4][r] + gz1[r]);
        nv = tanhf(gi[j1 + 8][r] + rr * gn1[r]);
        hreg[j1][r] = (1.0f - zz) * nv + zz * hreg[j1][r];
      }
    }
    __syncthreads();
    // publish new h to LDS in (m,k) layout for next step's A-fragments
    #pragma unroll
    for (int jt = 0; jt < 4; ++jt)
      #pragma unroll
      for (int r = 0; r < 8; ++r) {
        int m = r + ((l >> 4) * 8);
        int k = jt * 16 + (l & 15);
        sT[w][m * ST + k] = hreg[jt][r];
      }
    __syncthreads();
  }

  // global_mean_pool partial sums + counts
  #pragma unroll
  for (int jt = 0; jt < 4; ++jt) {
    int j = jt * 16 + (l & 15);
    #pragma unroll
    for (int r = 0; r < 8; ++r) {
      int m = r + ((l >> 4) * 8);
      int node = nodeBase + m;
      if (node < N_NODES) {
        int g = batch[node];
        atomicAdd(&sums[(long)g * H + j], hreg[jt][r]);
        if (jt == 0 && (l & 15) == 0) atomicAdd(&counts[g], 1.0f);
      }
    }
  }
}

// ---------------------------------------------------------------------------
// Kernel 3: readout: pooled = sums/max(counts,1); Linear(64->64)+ReLU;
// Linear(64->1). One 64-thread block per graph (tiny).
// ---------------------------------------------------------------------------
__global__ __launch_bounds__(64) void k_readout(
    const float* __restrict__ sums, const float* __restrict__ counts,
    const float* __restrict__ rW1, const float* __restrict__ rb1,
    const float* __restrict__ rW2, const float* __restrict__ rb2,
    float* __restrict__ out)
{
  __shared__ float pooled[H];
  __shared__ float a1[H];
  const int g = blockIdx.x;
  const int j = threadIdx.x;

  float cnt = fmaxf(counts[g], 1.0f);
  pooled[j] = sums[(long)g * H + j] / cnt;
  __syncthreads();

  float acc = rb1[j];
  #pragma unroll
  for (int k = 0; k < H; ++k) acc += pooled[k] * rW1[j * H + k];
  a1[j] = fmaxf(acc, 0.0f) * rW2[j];   // fold final weight into reduction
  __syncthreads();

  for (int s = 32; s > 0; s >>= 1) {
    if (j < s) a1[j] += a1[j + s];
    __syncthreads();
  }
  if (j == 0) out[g] = a1[0] + rb2[0];
}

// ---------------------------------------------------------------------------
extern "C" void kernel_launch(void* const* d_in, const int* in_sizes, int n_in,
                              void* d_out, int out_size, void* d_ws, size_t ws_size,
                              hipStream_t stream) {
  const float* x     = (const float*)d_in[0];
  const int*   ei    = (const int*)  d_in[1];
  const float* ea    = (const float*)d_in[2];
  const int*   batch = (const int*)  d_in[3];
  const float* mW1   = (const float*)d_in[4];
  const float* mb1   = (const float*)d_in[5];
  const float* mW2   = (const float*)d_in[6];
  const float* mb2   = (const float*)d_in[7];
  const float* Wih   = (const float*)d_in[8];
  const float* Whh   = (const float*)d_in[9];
  const float* bih   = (const float*)d_in[10];
  const float* bhh   = (const float*)d_in[11];
  const float* rW1   = (const float*)d_in[12];
  const float* rb1   = (const float*)d_in[13];
  const float* rW2   = (const float*)d_in[14];
  const float* rb2   = (const float*)d_in[15];
  float* out = (float*)d_out;

  // ws layout (floats): agg[N,H] | sums[G,H] | counts[G] | WihT | WhhT
  float* agg    = (float*)d_ws;
  float* sums   = agg + (long)N_NODES * H;
  float* counts = sums + (long)NGRAPH * H;
  float* WihT   = counts + NGRAPH;
  float* WhhT   = WihT + THREE_H * H;
  const long ztotal = (long)N_NODES * H + (long)NGRAPH * H + NGRAPH;

  k_zero<<<2048, 256, 0, stream>>>(agg, ztotal);
  k_repack<<<(THREE_H * H + 255) / 256, 256, 0, stream>>>(Wih, WihT);
  k_repack<<<(THREE_H * H + 255) / 256, 256, 0, stream>>>(Whh, WhhT);

  // 1.6M edges / (8 waves * 16 edges) = 12500 blocks, exact fit
  k_edge_mlp<<<N_EDGES / (8 * 16), 256, 0, stream>>>(
      x, ei, ea, mW1, mb1, mW2, mb2, agg);

  k_gru_pool<<<(N_NODES + 127) / 128, 256, 0, stream>>>(
      agg, batch, WihT, WhhT, bih, bhh, sums, counts);

  k_readout<<<NGRAPH, 64, 0, stream>>>(sums, counts, rW1, rb1, rW2, rb2, out);
}